// LatticePositionalEncoding_84963043049656
// MI455X (gfx1250) — compile-verified
//
#include <hip/hip_runtime.h>
#include <math.h>

typedef float v2f __attribute__((ext_vector_type(2)));
typedef float v4f __attribute__((ext_vector_type(4)));
typedef float v8f __attribute__((ext_vector_type(8)));

namespace {
constexpr int   kDHalf   = 512;
constexpr int   kDModel  = 1024;
constexpr int   kNPos    = 16 * 8192;   // B*S
constexpr int   kWaves   = 4;           // waves per block
constexpr int   kLdsRow  = 516;         // 512 + 4 pad floats (16B-aligned rows, de-strided banks)
constexpr float kEps     = 1e-5f;
}

__device__ __forceinline__ float gelu_exact(float x) {
    // torch-default exact GELU: 0.5 x (1 + erf(x / sqrt(2)))
    return 0.5f * x * (1.0f + erff(x * 0.70710678118654752f));
}

__device__ __forceinline__ void store_nt4(float* p, v4f v) {
    // NT store: output is streamed once (512 MB); keep L2 for the pe gather table.
    __builtin_nontemporal_store(v, (v4f*)p);
}

extern "C" __global__ __launch_bounds__(128, 1)
void lattice_pe_wmma_kernel(const int*   __restrict__ positions,
                            const float* __restrict__ pe,
                            const float* __restrict__ W1,     // [512,3] row-major
                            const float* __restrict__ b1,     // [512]
                            const float* __restrict__ gamma_, // [512]
                            const float* __restrict__ beta_,  // [512]
                            float*       __restrict__ out)    // [B*S, 1024]
{
    extern __shared__ float smem[];                  // kWaves * 16 * kLdsRow floats
    const int lane  = threadIdx.x & 31;
    const int wave  = threadIdx.x >> 5;
    const int tile  = blockIdx.x * kWaves + wave;    // 16 positions per wave-tile
    const int tileBase = tile * 16;
    const bool hi   = lane >= 16;
    const int  n    = lane & 15;                     // position-in-tile this lane owns

    // -------- lattice features for position n (spine is tiny & constant: unrolled) --------
    const int pos = positions[tileBase + n];
    constexpr int SP[10] = {0, 2, 4, 12, 36, 104, 304, 888, 2592, 7568};
    int level = 0, prev = 0, nxt = 0;
#pragma unroll
    for (int i = 0; i < 10; ++i) level += (SP[i] <= pos) ? 1 : 0;  // searchsorted 'right'
#pragma unroll
    for (int i = 0; i < 10; ++i) prev = (SP[i] <= pos) ? SP[i] : prev;
#pragma unroll
    for (int i = 9; i >= 0; --i) if (SP[i] > pos) nxt = SP[i];     // smallest spine > pos
    const float fLeft  = (float)(pos - prev);
    const float fRight = (nxt > 0) ? (float)(nxt - pos) : 0.0f;
    const float fLevel = (float)level;

    // -------- absolute-encoding half: gather pe rows (L2-resident), coalesced float4 copy --------
#pragma unroll 4
    for (int m = 0; m < 16; ++m) {
        const int pm = __shfl(pos, m, 32);
        const v4f* src = (const v4f*)(pe + (size_t)pm * kDHalf);
        float*     dst = out + (size_t)(tileBase + m) * kDModel;
#pragma unroll
        for (int c = 0; c < 4; ++c)
            store_nt4(dst + 4 * (lane + 32 * c), src[lane + 32 * c]);
    }

    // -------- lattice half: D = A(W1 16x4) x B(featsT 4x16) + C(b1) via WMMA f32 16x16x4 --------
    // B (4x16, K x N): VGPR0 = K0 | K2 halves, VGPR1 = K1 | K3 halves. K3 row = 0 pad.
    v2f bmat;
    bmat[0] = hi ? fLevel : fLeft;    // K=2 : K=0
    bmat[1] = hi ? 0.0f   : fRight;   // K=3 : K=1

    float* stage = smem + wave * (16 * kLdsRow);
    v8f acc   = {0.f, 0.f, 0.f, 0.f, 0.f, 0.f, 0.f, 0.f};  // vector LN partials: no per-iter
    v8f accsq = {0.f, 0.f, 0.f, 0.f, 0.f, 0.f, 0.f, 0.f};  // horizontal reduce / mov storm
    const int hioff = hi ? 8 : 0;
    const int kOff  = hi ? 2 : 0;

#pragma unroll 4
    for (int t = 0; t < 32; ++t) {
        const int dBase = t * 16;
        const int d     = dBase + n;  // this lane's A-matrix row (dim index)
        // A (16x4, M x K): VGPR0 = K0 | K2, VGPR1 = K1 | K3.
        // Load both words unconditionally (always in-bounds) -> select, no EXEC divergence.
        const float wA = W1[d * 3 + kOff];
        const float wB = W1[d * 3 + 1];
        v2f amat;
        amat[0] = wA;
        amat[1] = hi ? 0.0f : wB;
        // C = b1 tile in the 8-VGPR C/D layout (lane-half picks dims +0 / +8)
        const v4f c0 = *(const v4f*)(b1 + dBase + hioff);
        const v4f c1 = *(const v4f*)(b1 + dBase + hioff + 4);
        v8f cmat;
        cmat[0] = c0.x; cmat[1] = c0.y; cmat[2] = c0.z; cmat[3] = c0.w;
        cmat[4] = c1.x; cmat[5] = c1.y; cmat[6] = c1.z; cmat[7] = c1.w;

        v8f dacc = __builtin_amdgcn_wmma_f32_16x16x4_f32(
            /*neg_a=*/false, amat, /*neg_b=*/false, bmat,
            /*c_mod=*/(short)0, cmat, /*reuse_a=*/false, /*reuse_b=*/false);

        // D layout: VGPR r, lane n (lo half) = [dim dBase+r, pos n]; hi half = dim dBase+8+r.
        // => lane L holds 8 consecutive dims of position n: stage with 2x ds_store_b128.
        float* wp = stage + n * kLdsRow + dBase + hioff;
        v4f lo4, hi4;
        lo4.x = dacc[0]; lo4.y = dacc[1]; lo4.z = dacc[2]; lo4.w = dacc[3];
        hi4.x = dacc[4]; hi4.y = dacc[5]; hi4.z = dacc[6]; hi4.w = dacc[7];
        *(v4f*)(wp)     = lo4;
        *(v4f*)(wp + 4) = hi4;

        acc   = acc + dacc;           // 4x v_pk_add_f32
        accsq = accsq + dacc * dacc;  // 4x v_pk_fma_f32
    }

    // horizontal reduce of the 8-wide partials (once, outside the loop)
    float s1 = 0.0f, s2 = 0.0f;
#pragma unroll
    for (int r = 0; r < 8; ++r) { s1 += acc[r]; s2 += accsq[r]; }

    // combine lane-halves: lane n + lane n+16 together cover all 512 dims of position n
    s1 += __shfl_xor(s1, 16, 32);
    s2 += __shfl_xor(s2, 16, 32);
    const float mu  = s1 * (1.0f / 512.0f);
    const float var = s2 * (1.0f / 512.0f) - mu * mu;
    const float rs  = rsqrtf(var + kEps);

    __syncthreads();   // order DS stores before cross-lane DS reads

    // -------- fused LN + GELU + coalesced NT writeback from LDS --------
    v4f g4[4], be4[4];
#pragma unroll
    for (int it = 0; it < 4; ++it) {
        const int d0 = it * 128 + lane * 4;
        g4[it]  = *(const v4f*)(gamma_ + d0);
        be4[it] = *(const v4f*)(beta_  + d0);
    }

#pragma unroll 2
    for (int m = 0; m < 16; ++m) {
        const float muM = __shfl(mu, m, 32);
        const float rsM = __shfl(rs, m, 32);
        const float* srow = stage + m * kLdsRow;
        float*       orow = out + (size_t)(tileBase + m) * kDModel + kDHalf;
#pragma unroll
        for (int it = 0; it < 4; ++it) {
            const int d0 = it * 128 + lane * 4;
            const v4f h = *(const v4f*)(srow + d0);
            v4f r;
            r.x = gelu_exact((h.x - muM) * rsM * g4[it].x + be4[it].x);
            r.y = gelu_exact((h.y - muM) * rsM * g4[it].y + be4[it].y);
            r.z = gelu_exact((h.z - muM) * rsM * g4[it].z + be4[it].z);
            r.w = gelu_exact((h.w - muM) * rsM * g4[it].w + be4[it].w);
            store_nt4(orow + d0, r);
        }
    }
}

extern "C" void kernel_launch(void* const* d_in, const int* in_sizes, int n_in,
                              void* d_out, int out_size, void* d_ws, size_t ws_size,
                              hipStream_t stream) {
    (void)in_sizes; (void)n_in; (void)out_size; (void)d_ws; (void)ws_size;
    const int*   positions = (const int*)  d_in[0];
    const float* pe        = (const float*)d_in[1];
    const float* W1        = (const float*)d_in[2];
    const float* b1        = (const float*)d_in[3];
    const float* gamma_    = (const float*)d_in[4];
    const float* beta_     = (const float*)d_in[5];
    float* out = (float*)d_out;

    const int nTiles  = kNPos / 16;        // 8192 wave-tiles
    const int nBlocks = nTiles / kWaves;   // 2048 blocks of 128 threads (4 waves)
    const size_t shmem = (size_t)kWaves * 16 * kLdsRow * sizeof(float); // ~129 KB / WGP
    lattice_pe_wmma_kernel<<<nBlocks, 128, shmem, stream>>>(
        positions, pe, W1, b1, gamma_, beta_, out);
}